// HeteroGCN_59854664237669
// MI455X (gfx1250) — compile-verified
//
#include <hip/hip_runtime.h>
#include <hip/hip_bf16.h>

typedef float v2f __attribute__((ext_vector_type(2)));
typedef float v8f __attribute__((ext_vector_type(8)));

constexpr int NSPOT = 30000;
constexpr int NUSER = 30000;
constexpr int SPLITS = 5;
constexpr int AIN = 512;
constexpr int HEADS = 4;
constexpr int PER = 128;
constexpr int HID = 256;

// ---------------- utility kernels ----------------

__global__ void fill0_kernel(float* __restrict__ p, size_t n) {
    size_t i = (size_t)blockIdx.x * blockDim.x + threadIdx.x;
    if (i < n) p[i] = 0.0f;
}

__global__ void scale_kernel(float* __restrict__ p, float s, size_t n) {
    size_t i = (size_t)blockIdx.x * blockDim.x + threadIdx.x;
    if (i < n) p[i] *= s;
}

__global__ void deg_kernel(const int* __restrict__ edge, int E,
                           float* __restrict__ deg_s, float* __restrict__ deg_d) {
    int e = blockIdx.x * blockDim.x + threadIdx.x;
    if (e >= E) return;
    atomicAdd(&deg_s[edge[e]], 1.0f);
    atomicAdd(&deg_d[edge[E + e]], 1.0f);
}

__global__ void rdiv_kernel(const int* __restrict__ edge, int E,
                            const float* __restrict__ deg_s, const float* __restrict__ deg_d,
                            float* __restrict__ rdiv) {
    int e = blockIdx.x * blockDim.x + threadIdx.x;
    if (e >= E) return;
    rdiv[e] = rsqrtf(deg_s[edge[e]] * deg_d[edge[E + e]]);
}

// wq[h][i] = sum_o W_att[h][i][o] * q[h][o]   (collapses attention logit to x . wq)
__global__ void wqpack_kernel(const float* __restrict__ W_att, const float* __restrict__ q,
                              float* __restrict__ wq) {
    int t = blockIdx.x * blockDim.x + threadIdx.x;  // 0..2047
    if (t >= HEADS * AIN) return;
    int h = t >> 9, i = t & 511;
    const float* wrow = W_att + ((size_t)h * AIN + i) * PER;
    const float* qh = q + h * PER;
    float s = 0.0f;
    for (int o = 0; o < PER; ++o) s += wrow[o] * qh[o];
    wq[t] = s;
}

// one wave per node: 20 logits (5 splits x 4 heads), leaky-relu, softmax over splits
__global__ __launch_bounds__(256) void att_kernel(const float* __restrict__ X,
                                                  const float* __restrict__ wq,
                                                  float* __restrict__ att, int n_spot) {
    __shared__ float wql[HEADS * AIN];
    for (int i = threadIdx.x; i < HEADS * AIN; i += blockDim.x) wql[i] = wq[i];
    __syncthreads();
    int wid = threadIdx.x >> 5, lane = threadIdx.x & 31;
    int n = blockIdx.x * 8 + wid;
    if (n >= n_spot) return;
    const float* xr = X + (size_t)n * (SPLITS * AIN);
    float acc[SPLITS][HEADS];
#pragma unroll
    for (int s = 0; s < SPLITS; ++s)
#pragma unroll
        for (int h = 0; h < HEADS; ++h) acc[s][h] = 0.0f;
#pragma unroll
    for (int s = 0; s < SPLITS; ++s) {
        for (int j = 0; j < 16; ++j) {
            float xv = xr[s * AIN + j * 32 + lane];
#pragma unroll
            for (int h = 0; h < HEADS; ++h) acc[s][h] += xv * wql[h * AIN + j * 32 + lane];
        }
    }
#pragma unroll
    for (int s = 0; s < SPLITS; ++s)
#pragma unroll
        for (int h = 0; h < HEADS; ++h) {
            float v = acc[s][h];
            for (int m = 16; m > 0; m >>= 1) v += __shfl_xor(v, m, 32);
            acc[s][h] = v;
        }
    float a[SPLITS][HEADS];
#pragma unroll
    for (int h = 0; h < HEADS; ++h) {
        float l[SPLITS];
        float mx = -3.0e38f;
#pragma unroll
        for (int s = 0; s < SPLITS; ++s) {
            float v = acc[s][h];
            l[s] = v > 0.0f ? v : 0.2f * v;
            mx = fmaxf(mx, l[s]);
        }
        float sum = 0.0f;
#pragma unroll
        for (int s = 0; s < SPLITS; ++s) { l[s] = __expf(l[s] - mx); sum += l[s]; }
        float inv = 1.0f / sum;
#pragma unroll
        for (int s = 0; s < SPLITS; ++s) a[s][h] = l[s] * inv;
    }
    if (lane == 0) {
        float* ap = att + (size_t)n * (SPLITS * HEADS);
#pragma unroll
        for (int s = 0; s < SPLITS; ++s)
#pragma unroll
            for (int h = 0; h < HEADS; ++h) ap[s * HEADS + h] = a[s][h];
    }
}

// Y[n,:] = sum_s att[n,s,h] * X[n,s,:]   (head h)
__global__ void mix_kernel(const float* __restrict__ X, const float* __restrict__ att,
                           float* __restrict__ Y, int h, int n_spot) {
    size_t idx = (size_t)blockIdx.x * blockDim.x + threadIdx.x;
    size_t total = (size_t)n_spot * (AIN / 4);
    if (idx >= total) return;
    int n = (int)(idx >> 7);
    int c4 = (int)(idx & 127);
    const float4* xr = (const float4*)X + (size_t)n * (SPLITS * AIN / 4) + c4;
    const float* ap = att + (size_t)n * (SPLITS * HEADS) + h;
    float4 acc = {0.f, 0.f, 0.f, 0.f};
#pragma unroll
    for (int s = 0; s < SPLITS; ++s) {
        float w = ap[s * HEADS];
        float4 xv = xr[s * (AIN / 4)];
        acc.x += w * xv.x; acc.y += w * xv.y; acc.z += w * xv.z; acc.w += w * xv.w;
    }
    ((float4*)Y)[(size_t)n * (AIN / 4) + c4] = acc;
}

// ---------------- fp32 WMMA GEMM: C[M,N] = A[M,K] @ B[K,N] ----------------
// block = 256 threads (8 waves), tile 128x64, K staged 16-wide in LDS.
// Requires: K % 16 == 0, N % 64 == 0 (M guarded).
#define BM 128
#define BN 64
#define KT 16

__global__ __launch_bounds__(256) void gemm_f32_kernel(const float* __restrict__ A, int lda,
                                                       const float* __restrict__ B, int ldb,
                                                       float* __restrict__ C, int ldc,
                                                       int M, int N, int K) {
    __shared__ float As[BM * (KT + 1)];
    __shared__ float Bs[KT * BN];
    int tid = threadIdx.x;
    int m0 = blockIdx.x * BM;
    int n0 = blockIdx.y * BN;
    int wid = tid >> 5, lane = tid & 31;
    int wm = wid >> 1, wn = wid & 1;      // 4x2 wave grid, 32x32 per wave
    int half = lane >> 4, lrow = lane & 15;

    v8f zero = {0.f, 0.f, 0.f, 0.f, 0.f, 0.f, 0.f, 0.f};
    v8f acc[2][2];
    acc[0][0] = zero; acc[0][1] = zero; acc[1][0] = zero; acc[1][1] = zero;

    int nk = K / KT;
    for (int kt = 0; kt < nk; ++kt) {
        int k0 = kt * KT;
        // stage A tile 128x16 (512 float4s, 2 per thread), zero-padded rows past M
#pragma unroll
        for (int i = 0; i < 2; ++i) {
            int idx = tid + i * 256;           // 0..511
            int row = idx >> 2;
            int kc = (idx & 3) << 2;
            float4 v = {0.f, 0.f, 0.f, 0.f};
            int grow = m0 + row;
            if (grow < M) v = *(const float4*)(A + (size_t)grow * lda + k0 + kc);
            int o = row * (KT + 1) + kc;
            As[o] = v.x; As[o + 1] = v.y; As[o + 2] = v.z; As[o + 3] = v.w;
        }
        // stage B tile 16x64 (256 float4s, 1 per thread)
        {
            int krow = tid >> 4;
            int cc = (tid & 15) << 2;
            float4 v = *(const float4*)(B + (size_t)(k0 + krow) * ldb + n0 + cc);
            int o = krow * BN + cc;
            Bs[o] = v.x; Bs[o + 1] = v.y; Bs[o + 2] = v.z; Bs[o + 3] = v.w;
        }
        __syncthreads();
#pragma unroll
        for (int kb = 0; kb < 4; ++kb) {
            int kk = kb * 4 + half * 2;        // lanes 0-15: K{0,1}, lanes 16-31: K{2,3}
            v2f a[2], b[2];
#pragma unroll
            for (int i = 0; i < 2; ++i) {
                int row = wm * 32 + i * 16 + lrow;
                a[i].x = As[row * (KT + 1) + kk];
                a[i].y = As[row * (KT + 1) + kk + 1];
            }
#pragma unroll
            for (int j = 0; j < 2; ++j) {
                int col = wn * 32 + j * 16 + lrow;
                b[j].x = Bs[kk * BN + col];
                b[j].y = Bs[(kk + 1) * BN + col];
            }
#pragma unroll
            for (int i = 0; i < 2; ++i)
#pragma unroll
                for (int j = 0; j < 2; ++j)
                    acc[i][j] = __builtin_amdgcn_wmma_f32_16x16x4_f32(
                        false, a[i], false, b[j], (short)0, acc[i][j], false, false);
        }
        __syncthreads();
    }
    // C layout: VGPR r -> row (half*8 + r), lane 0-15 cols
#pragma unroll
    for (int i = 0; i < 2; ++i) {
        int rbase = m0 + wm * 32 + i * 16 + half * 8;
#pragma unroll
        for (int j = 0; j < 2; ++j) {
            int col = n0 + wn * 32 + j * 16 + lrow;
#pragma unroll
            for (int r = 0; r < 8; ++r) {
                int grow = rbase + r;
                if (grow < M) C[(size_t)grow * ldc + col] = acc[i][j][r];
            }
        }
    }
}

// ---------------- edge scatter: out[dst] += msg[src] * rdiv[e] ----------------
__global__ __launch_bounds__(256) void scatter_kernel(const int* __restrict__ edge, int E,
                                                      const float* __restrict__ rdiv,
                                                      const float* __restrict__ msg,
                                                      float* __restrict__ out) {
    int e = blockIdx.x * 4 + (threadIdx.x >> 6);
    int q = threadIdx.x & 63;
    if (e >= E) return;
    int s = edge[e], d = edge[E + e];
    float w = rdiv[e];
    float4 v = ((const float4*)msg)[(size_t)s * 64 + q];
    float* o = out + (size_t)d * HID + q * 4;
    atomicAdd(o + 0, v.x * w);
    atomicAdd(o + 1, v.y * w);
    atomicAdd(o + 2, v.z * w);
    atomicAdd(o + 3, v.w * w);
}

__global__ void relu_acc_kernel(float* __restrict__ x, float* __restrict__ acc, size_t n) {
    size_t i = (size_t)blockIdx.x * blockDim.x + threadIdx.x;
    if (i >= n) return;
    float v = x[i];
    v = v > 0.0f ? v : 0.0f;
    x[i] = v;
    acc[i] += v;
}

// out[n] = x[n,:] . W + b   (wave per node)
__global__ __launch_bounds__(256) void headout_kernel(const float* __restrict__ x,
                                                      const float* __restrict__ W,
                                                      const float* __restrict__ b,
                                                      float* __restrict__ out, int n_nodes) {
    int wid = threadIdx.x >> 5, lane = threadIdx.x & 31;
    int n = blockIdx.x * 8 + wid;
    if (n >= n_nodes) return;
    const float* xr = x + (size_t)n * HID;
    float v = 0.0f;
#pragma unroll
    for (int j = 0; j < 8; ++j) v += xr[j * 32 + lane] * W[j * 32 + lane];
    for (int m = 16; m > 0; m >>= 1) v += __shfl_xor(v, m, 32);
    if (lane == 0) out[n] = v + b[0];
}

// ---------------- host orchestration ----------------

static inline int cdiv(int a, int b) { return (a + b - 1) / b; }

extern "C" void kernel_launch(void* const* d_in, const int* in_sizes, int n_in,
                              void* d_out, int out_size, void* d_ws, size_t ws_size,
                              hipStream_t stream) {
    (void)n_in; (void)out_size; (void)ws_size;
    const float* x_spot = (const float*)d_in[0];
    const float* x_user = (const float*)d_in[1];
    const int* edge_us = (const int*)d_in[2];
    const int* edge_su = (const int*)d_in[3];
    const float* W_att = (const float*)d_in[4];
    const float* q_att = (const float*)d_in[5];
    const float* W0_us = (const float*)d_in[6];
    const float* W0_su = (const float*)d_in[7];
    const float* Wmid_us = (const float*)d_in[8];
    const float* Wmid_su = (const float*)d_in[9];
    const float* W_out_s = (const float*)d_in[10];
    const float* b_out_s = (const float*)d_in[11];
    const float* W_out_u = (const float*)d_in[12];
    const float* b_out_u = (const float*)d_in[13];
    const int E = in_sizes[2] / 2;
    float* out = (float*)d_out;

    // workspace carve-up (floats)
    float* ws = (float*)d_ws;
    size_t off = 0;
    float* rdiv_us = ws + off; off += 600064;
    float* rdiv_su = ws + off; off += 600064;
    float* deg = ws + off;     off += 122880;                   // 4 x 30000 (+pad)
    float* att = ws + off;     off += 600064;                   // [N,5,4]
    float* wq = ws + off;      off += 4096;                     // [4,512]
    float* Y = ws + off;       off += (size_t)NSPOT * 512;      // mix buffer (aliased as msgs)
    float* xs512 = ws + off;   off += (size_t)NSPOT * 512;      // attention output
    float* sbuf0 = ws + off;   off += (size_t)NSPOT * HID;
    float* sbuf1 = ws + off;   off += (size_t)NSPOT * HID;
    float* ubuf0 = ws + off;   off += (size_t)NUSER * HID;
    float* ubuf1 = ws + off;   off += (size_t)NUSER * HID;
    float* msg_u = Y;                                   // [NUSER,256]
    float* msg_s = Y + (size_t)NUSER * HID;             // [NSPOT,256]

    // 1) symmetric GCN normalization
    fill0_kernel<<<cdiv(4 * NSPOT, 256), 256, 0, stream>>>(deg, (size_t)4 * NSPOT);
    deg_kernel<<<cdiv(E, 256), 256, 0, stream>>>(edge_us, E, deg, deg + NSPOT);
    deg_kernel<<<cdiv(E, 256), 256, 0, stream>>>(edge_su, E, deg + 2 * NSPOT, deg + 3 * NSPOT);
    rdiv_kernel<<<cdiv(E, 256), 256, 0, stream>>>(edge_us, E, deg, deg + NSPOT, rdiv_us);
    rdiv_kernel<<<cdiv(E, 256), 256, 0, stream>>>(edge_su, E, deg + 2 * NSPOT, deg + 3 * NSPOT, rdiv_su);

    // 2) attention: logits via x.(W@q), softmax over splits, mix, then per-head WMMA GEMM
    wqpack_kernel<<<8, 256, 0, stream>>>(W_att, q_att, wq);
    att_kernel<<<cdiv(NSPOT, 8), 256, 0, stream>>>(x_spot, wq, att, NSPOT);
    for (int h = 0; h < HEADS; ++h) {
        mix_kernel<<<cdiv(NSPOT * (AIN / 4), 256), 256, 0, stream>>>(x_spot, att, Y, h, NSPOT);
        gemm_f32_kernel<<<dim3(cdiv(NSPOT, BM), PER / BN), 256, 0, stream>>>(
            Y, AIN, W_att + (size_t)h * AIN * PER, PER, xs512 + (size_t)h * PER, AIN,
            NSPOT, PER, AIN);
    }

    // 3) zero the layer-mean accumulators living in d_out
    fill0_kernel<<<cdiv(2 * NSPOT * HID, 256), 256, 0, stream>>>(out, (size_t)2 * NSPOT * HID);

    // 4) GCN layers
    const float* cs = xs512; int ks = AIN;
    const float* cu = x_user; int ku = AIN;
    float* sb[2] = {sbuf0, sbuf1};
    float* ub[2] = {ubuf0, ubuf1};
    for (int l = 0; l < 3; ++l) {
        const float* Wus = (l == 0) ? W0_us : Wmid_us + (size_t)(l - 1) * HID * HID;
        const float* Wsu = (l == 0) ? W0_su : Wmid_su + (size_t)(l - 1) * HID * HID;
        gemm_f32_kernel<<<dim3(cdiv(NUSER, BM), HID / BN), 256, 0, stream>>>(
            cu, ku, Wus, HID, msg_u, HID, NUSER, HID, ku);
        gemm_f32_kernel<<<dim3(cdiv(NSPOT, BM), HID / BN), 256, 0, stream>>>(
            cs, ks, Wsu, HID, msg_s, HID, NSPOT, HID, ks);
        float* ns = sb[l & 1];
        float* nu = ub[l & 1];
        fill0_kernel<<<cdiv(NSPOT * HID, 256), 256, 0, stream>>>(ns, (size_t)NSPOT * HID);
        fill0_kernel<<<cdiv(NUSER * HID, 256), 256, 0, stream>>>(nu, (size_t)NUSER * HID);
        scatter_kernel<<<cdiv(E, 4), 256, 0, stream>>>(edge_us, E, rdiv_us, msg_u, ns);
        scatter_kernel<<<cdiv(E, 4), 256, 0, stream>>>(edge_su, E, rdiv_su, msg_s, nu);
        relu_acc_kernel<<<cdiv(NSPOT * HID, 256), 256, 0, stream>>>(ns, out, (size_t)NSPOT * HID);
        relu_acc_kernel<<<cdiv(NUSER * HID, 256), 256, 0, stream>>>(
            nu, out + (size_t)NSPOT * HID, (size_t)NUSER * HID);
        cs = ns; cu = nu; ks = HID; ku = HID;
    }

    // 5) means + output heads
    scale_kernel<<<cdiv(2 * NSPOT * HID, 256), 256, 0, stream>>>(out, 1.0f / 3.0f,
                                                                 (size_t)2 * NSPOT * HID);
    float* out_heads = out + (size_t)2 * NSPOT * HID;
    headout_kernel<<<cdiv(NSPOT, 8), 256, 0, stream>>>(cs, W_out_s, b_out_s, out_heads, NSPOT);
    headout_kernel<<<cdiv(NUSER, 8), 256, 0, stream>>>(cu, W_out_u, b_out_u,
                                                       out_heads + NSPOT, NUSER);
}